// CPC_NCE_Loss_91285234909676
// MI455X (gfx1250) — compile-verified
//
#include <hip/hip_runtime.h>
#include <math.h>

typedef __attribute__((ext_vector_type(2))) float v2f;
typedef __attribute__((ext_vector_type(8))) float v8f;

#define BB 256       // batch (b and c)
#define DD 128       // feature dim (K)
#define TFULL 2049   // time dim of pred
#define TT 2048      // valid t positions
#define BM 64        // b-rows per workgroup
#define TCHUNK 32    // consecutive t per workgroup -> full cacheline use per (row,d)
#define LDA 130      // padded LDS row stride (floats): conflict-free frag loads, 8B aligned
#define LDB 130
#define LDG 260

#define A_FLOATS (BM * LDA)   // 8320
#define B_FLOATS (BB * LDB)   // 33280  (G: 64*260=16640 floats aliases this region)

// gfx1250 async memory->LDS: per-lane 4B transfer, tracked by ASYNCcnt.
// VDST vgpr = LDS byte address (low 32 bits of flat __shared__ pointer),
// VADDR pair = 64-bit global address.
__device__ __forceinline__ void async_ld_f32(unsigned lds_byte, const float* gptr) {
    asm volatile("global_load_async_to_lds_b32 %0, %1, off"
                 :
                 : "v"(lds_byte), "v"(gptr)
                 : "memory");
}
__device__ __forceinline__ void wait_async0() {
#if __has_builtin(__builtin_amdgcn_s_wait_asynccnt)
    __builtin_amdgcn_s_wait_asynccnt(0);
#else
    asm volatile("s_wait_asynccnt 0" ::: "memory");
#endif
}

__global__ __launch_bounds__(256, 1)
void cpc_nce_main(const float* __restrict__ pred, float* __restrict__ part_out) {
    __shared__ float smem[A_FLOATS + B_FLOATS];   // 162.5 KB of the 320 KB WGP LDS
    __shared__ float red[256];
    float* As = smem;
    float* Bs = smem + A_FLOATS;
    float* Gs = Bs;                                // alias: G written after all B reads done

    const unsigned As_base = (unsigned)(uintptr_t)As;  // LDS byte offset (flat addr low bits)
    const unsigned Bs_base = (unsigned)(uintptr_t)Bs;

    const int tid  = threadIdx.x;
    const int wave = tid >> 5;
    const int lane = tid & 31;
    const int mt   = wave & 3;    // m-tile (16 rows) within the 64-row block
    const int nh   = wave >> 2;   // which half of the 16 n-tiles
    const int mrow = lane & 15;   // row for A frag / col for B frag / col for C,D
    const int kh   = lane >> 4;   // K half-select per ISA 16x4 f32 A layout

    const int b0 = blockIdx.y * BM;
    const int t0 = blockIdx.x * TCHUNK;

    const size_t SB = (size_t)DD * TFULL;  // b stride in elements
    const size_t SD = (size_t)TFULL;       // d stride in elements

    float local_acc = 0.f;

    for (int ti = 0; ti < TCHUNK; ++ti) {
        const int t = t0 + ti;

        // ---- async-stage A = ht[b0:b0+64, :, t] : 8192 elems, 32/thread ----
        #pragma unroll 8
        for (int i = 0; i < (BM * DD) / 256; ++i) {
            int idx = tid + i * 256;
            int b = idx >> 7, d = idx & 127;
            async_ld_f32(As_base + (unsigned)(b * LDA + d) * 4u,
                         pred + (size_t)(b0 + b) * SB + (size_t)d * SD + t);
        }
        // ---- async-stage B = ht_[:, :, t] = pred[:, :, t+1] : 32768 elems, 128/thread ----
        #pragma unroll 8
        for (int i = 0; i < (BB * DD) / 256; ++i) {
            int idx = tid + i * 256;
            int c = idx >> 7, d = idx & 127;
            async_ld_f32(Bs_base + (unsigned)(c * LDB + d) * 4u,
                         pred + (size_t)c * SB + (size_t)d * SD + (t + 1));
        }
        wait_async0();
        __syncthreads();

        // ---- G tile compute: 8 accumulators of 16x16 f32 per wave ----
        v8f acc[8];
        #pragma unroll
        for (int j = 0; j < 8; ++j) { v8f z = {}; acc[j] = z; }

        const float* arow = &As[(mt * 16 + mrow) * LDA + 2 * kh];
        const int    bbase = mrow * LDB + 2 * kh;

        #pragma unroll 4
        for (int kk = 0; kk < DD / 4; ++kk) {
            v2f a = *(const v2f*)(arow + kk * 4);          // A frag reused across 8 n-tiles
            #pragma unroll
            for (int j = 0; j < 8; ++j) {
                const float* bp = &Bs[((nh * 8 + j) * 16) * LDB + bbase + kk * 4];
                v2f b = *(const v2f*)bp;
                acc[j] = __builtin_amdgcn_wmma_f32_16x16x4_f32(
                    /*neg_a=*/false, a, /*neg_b=*/false, b,
                    /*c_mod=*/(short)0, acc[j],
                    /*reuse_a=*/false, /*reuse_b=*/false);
            }
        }
        __syncthreads();   // all Bs reads complete before Gs (alias) is written

        // ---- spill G to LDS per C/D layout: VGPR v -> row v (lanes 0-15) / v+8 (lanes 16-31) ----
        #pragma unroll
        for (int j = 0; j < 8; ++j) {
            int col = (nh * 8 + j) * 16 + mrow;
            #pragma unroll
            for (int v = 0; v < 8; ++v) {
                int row = mt * 16 + v + kh * 8;
                Gs[row * LDG + col] = acc[j][v];
            }
        }
        __syncthreads();

        // ---- log(sum_c exp(G[b,c] - G[b,b])): 4 threads per row ----
        {
            int row  = tid >> 2;
            int part = tid & 3;
            float fz = Gs[row * LDG + (b0 + row)];
            const float* g = &Gs[row * LDG + part * 64];
            float s = 0.f;
            #pragma unroll 8
            for (int c = 0; c < 64; ++c) s += expf(g[c] - fz);
            s += __shfl_xor(s, 1, 32);
            s += __shfl_xor(s, 2, 32);
            if (part == 0) local_acc += logf(s);
        }
        __syncthreads();   // Gs/As reused next iteration
    }

    // ---- block reduction of local log-sum terms ----
    red[tid] = local_acc;
    __syncthreads();
    #pragma unroll
    for (int s = 128; s > 0; s >>= 1) {
        if (tid < s) red[tid] += red[tid + s];
        __syncthreads();
    }
    if (tid == 0) part_out[blockIdx.y * gridDim.x + blockIdx.x] = red[0];
}

__global__ __launch_bounds__(256)
void cpc_nce_finish(const float* __restrict__ part, float* __restrict__ out,
                    int n, float inv_count) {
    __shared__ float red[256];
    float s = 0.f;
    for (int i = threadIdx.x; i < n; i += 256) s += part[i];
    red[threadIdx.x] = s;
    __syncthreads();
    #pragma unroll
    for (int k = 128; k > 0; k >>= 1) {
        if (threadIdx.x < k) red[threadIdx.x] += red[threadIdx.x + k];
        __syncthreads();
    }
    if (threadIdx.x == 0) out[0] = red[0] * inv_count;
}

extern "C" void kernel_launch(void* const* d_in, const int* in_sizes, int n_in,
                              void* d_out, int out_size, void* d_ws, size_t ws_size,
                              hipStream_t stream) {
    (void)in_sizes; (void)n_in; (void)out_size; (void)ws_size;
    const float* pred = (const float*)d_in[0];
    float* out  = (float*)d_out;
    float* part = (float*)d_ws;   // (2048/TCHUNK)*(256/BM) = 256 floats

    dim3 grid(TT / TCHUNK, BB / BM);   // (64, 4)
    cpc_nce_main<<<grid, 256, 0, stream>>>(pred, part);

    const int nwg = (TT / TCHUNK) * (BB / BM);
    cpc_nce_finish<<<1, 256, 0, stream>>>(part, out, nwg, 1.0f / ((float)TT * (float)BB));
}